// RGAT_22935125360817
// MI455X (gfx1250) — compile-verified
//
#include <hip/hip_runtime.h>
#include <math.h>

#define N_NODES 50000
#define N_EDGES 800000
#define IN_C    128
#define HC      128          // HEADS * OUT_C
#define R_REL   8
#define HEADS   4
#define NEG_SLOPE 0.2f

typedef float v2f __attribute__((ext_vector_type(2)));
typedef float v8f __attribute__((ext_vector_type(8)));

// ---------------------------------------------------------------------------
// Sign-aware float atomic max (int-ordering trick, valid with -inf init):
//   positive values: signed-int max; negative values: unsigned min.
// Cross-updates cannot corrupt: a stored positive always wins both orderings.
// ---------------------------------------------------------------------------
__device__ __forceinline__ void atomicMaxF(float* addr, float v) {
    if (v >= 0.0f) {
        atomicMax((int*)addr, __float_as_int(v));
    } else {
        atomicMin((unsigned int*)addr, __float_as_uint(v));
    }
}

// ---------------------------------------------------------------------------
// Kernel 0: init out = bias (folds "+bias" and empty-node case), amax = -inf,
// denom = 0.   Grid covers N*HC elements.
// ---------------------------------------------------------------------------
__global__ __launch_bounds__(256) void rgat_init(float* __restrict__ out,
                                                 const float* __restrict__ bias,
                                                 float* __restrict__ amax,
                                                 float* __restrict__ denom) {
    long i = (long)blockIdx.x * 256 + threadIdx.x;
    if (i < (long)N_NODES * HC) out[i] = bias[i & (HC - 1)];
    if (i < (long)N_NODES * HEADS) {
        amax[i]  = -__builtin_huge_valf();
        denom[i] = 0.0f;
    }
}

// ---------------------------------------------------------------------------
// Kernel 1: T[n, r*HC+o] = x[n,:] @ W[r][:,o]   via V_WMMA_F32_16X16X4_F32.
// One wave per (16-row tile, 256-col group). x tile staged in LDS (stride 132
// floats -> conflict-free column reads), A kept in 64 VGPRs across K=128.
// Two 16-col tiles processed per iteration: two independent WMMA accumulator
// chains share the A registers, doubling ILP and in-flight B loads.
// W (512 KB) is L2-resident; B chunks streamed per 16x16x4 step.
// ---------------------------------------------------------------------------
#define ASTRIDE 132

__global__ __launch_bounds__(32) void rgat_gemm(const float* __restrict__ x,
                                                const float* __restrict__ W,
                                                float* __restrict__ T) {
    const int row0 = blockIdx.x * 16;        // 3125 row tiles (50000 = 3125*16)
    const int cg   = blockIdx.y;             // 4 col groups of 256
    const int lane = threadIdx.x;

    __shared__ float As[16 * ASTRIDE];

    // Stage x tile 16x128 into LDS (coalesced b128 loads).
    {
        const float4* xv = reinterpret_cast<const float4*>(x + (long)row0 * IN_C);
        #pragma unroll
        for (int i = 0; i < 16; ++i) {
            float4 v = xv[i * 32 + lane];     // row i, cols 4*lane..4*lane+3
            *reinterpret_cast<float4*>(&As[i * ASTRIDE + lane * 4]) = v;
        }
    }
    __syncthreads();

    // A-matrix layout for WMMA f32 16x16x4 (ISA 7.12.2):
    //  lanes 0-15 : M=lane,    VGPR0=K(4c+0), VGPR1=K(4c+1)
    //  lanes16-31 : M=lane-16, VGPR0=K(4c+2), VGPR1=K(4c+3)
    const int mrow = lane & 15;
    const int koff = (lane >> 4) * 2;
    v2f a[32];
    #pragma unroll
    for (int c = 0; c < 32; ++c) {
        a[c].x = As[mrow * ASTRIDE + 4 * c + koff];
        a[c].y = As[mrow * ASTRIDE + 4 * c + koff + 1];
    }

    const int ncol = lane & 15;
    #pragma unroll 1
    for (int it = 0; it < 8; ++it) {
        const int j0 = cg * 256 + it * 32;       // output col base (0..1023)
        const int r  = j0 >> 7;                  // relation (32 | 128: no straddle)
        const int o0 = j0 & (HC - 1);            // col within relation
        const float* Wr = W + (long)r * IN_C * HC + o0 + ncol;

        v8f acc0 = {};
        v8f acc1 = {};
        #pragma unroll
        for (int c = 0; c < 32; ++c) {
            // B-matrix 4x16: lanes 0-15 rows K(4c+0)/K(4c+1),
            //                lanes16-31 rows K(4c+2)/K(4c+3)  (mirrors A split)
            const long rowo0 = (long)(4 * c + koff) * HC;
            const long rowo1 = (long)(4 * c + koff + 1) * HC;
            v2f b0, b1;
            b0.x = Wr[rowo0];
            b0.y = Wr[rowo1];
            b1.x = Wr[rowo0 + 16];
            b1.y = Wr[rowo1 + 16];
            acc0 = __builtin_amdgcn_wmma_f32_16x16x4_f32(
                false, a[c], false, b0, (short)0, acc0, false, false);
            acc1 = __builtin_amdgcn_wmma_f32_16x16x4_f32(
                false, a[c], false, b1, (short)0, acc1, false, false);
        }

        // D layout: VGPR r8 -> (M=r8, N=lane) lanes 0-15, (M=r8+8) lanes 16-31
        const int m0 = (lane >> 4) * 8;
        float* Tp = T + (long)(row0 + m0) * (R_REL * HC) + j0 + ncol;
        #pragma unroll
        for (int r8 = 0; r8 < 8; ++r8) {
            Tp[(long)r8 * (R_REL * HC)]      = acc0[r8];
            Tp[(long)r8 * (R_REL * HC) + 16] = acc1[r8];
        }
    }
}

// ---------------------------------------------------------------------------
// Kernel 2: per-edge attention logits + segment max.
// Wave per edge: coalesced float4 gathers of T rows; 4-head partials per lane;
// wave32 butterfly reduction; leaky-relu; atomic float max into amax[dst].
// ---------------------------------------------------------------------------
__global__ __launch_bounds__(256) void rgat_attn(const int* __restrict__ ei,
                                                 const int* __restrict__ et,
                                                 const float* __restrict__ T,
                                                 const float* __restrict__ q,
                                                 const float* __restrict__ kw,
                                                 float* __restrict__ alpha,
                                                 float* __restrict__ amax) {
    const int e = blockIdx.x * 8 + (threadIdx.x >> 5);
    if (e >= N_EDGES) return;
    const int lane = threadIdx.x & 31;

    const int s = ei[e];
    const int d = ei[N_EDGES + e];
    const int t = et[e];

    const float4* Ti = reinterpret_cast<const float4*>(T + ((long)d * R_REL + t) * HC);
    const float4* Tj = reinterpret_cast<const float4*>(T + ((long)s * R_REL + t) * HC);
    const float4 oi = Ti[lane];               // outi (dst) cols 4*lane..+3
    const float4 oj = Tj[lane];               // outj (src)

    // q,k are [HC,4] row-major: row o is one float4 of the 4 heads.
    const float4* q4 = reinterpret_cast<const float4*>(q)  + 4 * lane;
    const float4* k4 = reinterpret_cast<const float4*>(kw) + 4 * lane;

    float p0 = 0.f, p1 = 0.f, p2 = 0.f, p3 = 0.f;
    {
        float4 qq = q4[0], kk = k4[0];
        p0 += oi.x * qq.x + oj.x * kk.x;  p1 += oi.x * qq.y + oj.x * kk.y;
        p2 += oi.x * qq.z + oj.x * kk.z;  p3 += oi.x * qq.w + oj.x * kk.w;
    }
    {
        float4 qq = q4[1], kk = k4[1];
        p0 += oi.y * qq.x + oj.y * kk.x;  p1 += oi.y * qq.y + oj.y * kk.y;
        p2 += oi.y * qq.z + oj.y * kk.z;  p3 += oi.y * qq.w + oj.y * kk.w;
    }
    {
        float4 qq = q4[2], kk = k4[2];
        p0 += oi.z * qq.x + oj.z * kk.x;  p1 += oi.z * qq.y + oj.z * kk.y;
        p2 += oi.z * qq.z + oj.z * kk.z;  p3 += oi.z * qq.w + oj.z * kk.w;
    }
    {
        float4 qq = q4[3], kk = k4[3];
        p0 += oi.w * qq.x + oj.w * kk.x;  p1 += oi.w * qq.y + oj.w * kk.y;
        p2 += oi.w * qq.z + oj.w * kk.z;  p3 += oi.w * qq.w + oj.w * kk.w;
    }

    #pragma unroll
    for (int off = 16; off >= 1; off >>= 1) {
        p0 += __shfl_xor(p0, off, 32);
        p1 += __shfl_xor(p1, off, 32);
        p2 += __shfl_xor(p2, off, 32);
        p3 += __shfl_xor(p3, off, 32);
    }

    if (lane < HEADS) {
        float v = (lane == 0) ? p0 : (lane == 1) ? p1 : (lane == 2) ? p2 : p3;
        v = (v > 0.0f) ? v : NEG_SLOPE * v;          // leaky relu
        alpha[(long)e * HEADS + lane] = v;
        atomicMaxF(&amax[(long)d * HEADS + lane], v);
    }
}

// ---------------------------------------------------------------------------
// Kernel 3: ex = exp(alpha - amax[dst]); denom[dst] += ex (atomic).
// One thread per (edge, head).
// ---------------------------------------------------------------------------
__global__ __launch_bounds__(256) void rgat_softmax(const int* __restrict__ ei,
                                                    const float* __restrict__ amax,
                                                    float* __restrict__ alpha,
                                                    float* __restrict__ denom) {
    long i = (long)blockIdx.x * 256 + threadIdx.x;
    if (i >= (long)N_EDGES * HEADS) return;
    const int e = (int)(i >> 2);
    const int h = (int)(i & 3);
    const int d = ei[N_EDGES + e];
    const float ex = __expf(alpha[i] - amax[(long)d * HEADS + h]);
    alpha[i] = ex;
    atomicAdd(&denom[(long)d * HEADS + h], ex);
}

// ---------------------------------------------------------------------------
// Kernel 4: out[dst] += outj * (ex / (denom[dst]+1e-16)), scatter via f32
// atomics. Wave per edge; lane owns 4 contiguous channels of one head.
// ---------------------------------------------------------------------------
__global__ __launch_bounds__(256) void rgat_agg(const int* __restrict__ ei,
                                                const int* __restrict__ et,
                                                const float* __restrict__ T,
                                                const float* __restrict__ alpha,
                                                const float* __restrict__ denom,
                                                float* __restrict__ out) {
    const int e = blockIdx.x * 8 + (threadIdx.x >> 5);
    if (e >= N_EDGES) return;
    const int lane = threadIdx.x & 31;

    const int s = ei[e];
    const int d = ei[N_EDGES + e];
    const int t = et[e];

    const float4* Tj = reinterpret_cast<const float4*>(T + ((long)s * R_REL + t) * HC);
    const float4 oj = Tj[lane];

    const int h = lane >> 3;                       // head of channels 4*lane..
    const float w = alpha[(long)e * HEADS + h] /
                    (denom[(long)d * HEADS + h] + 1e-16f);

    float* op = out + (long)d * HC + lane * 4;
    atomicAdd(op + 0, oj.x * w);
    atomicAdd(op + 1, oj.y * w);
    atomicAdd(op + 2, oj.z * w);
    atomicAdd(op + 3, oj.w * w);
}

// ---------------------------------------------------------------------------
// Host-side launch
// ---------------------------------------------------------------------------
extern "C" void kernel_launch(void* const* d_in, const int* in_sizes, int n_in,
                              void* d_out, int out_size, void* d_ws, size_t ws_size,
                              hipStream_t stream) {
    const float* x    = (const float*)d_in[0];
    const int*   ei   = (const int*)  d_in[1];   // [2, E]
    const int*   et   = (const int*)  d_in[2];   // [E]
    const float* W    = (const float*)d_in[3];   // [R, IN_C, HC]
    const float* q    = (const float*)d_in[4];   // [HC, H]
    const float* kw   = (const float*)d_in[5];   // [HC, H]
    const float* bias = (const float*)d_in[6];   // [HC]
    float*       out  = (float*)d_out;           // [N, HC]

    // Workspace carve-up (all 256-B aligned by construction).
    char* ws = (char*)d_ws;
    float* T     = (float*)(ws);                                   // 204.8 MB
    float* alpha = (float*)(ws + (size_t)N_NODES * R_REL * HC * 4);// 12.8 MB
    float* amax  = (float*)((char*)alpha + (size_t)N_EDGES * HEADS * 4);
    float* denom = (float*)((char*)amax  + (size_t)N_NODES * HEADS * 4);

    // 0) init out=bias, amax=-inf, denom=0
    {
        long n = (long)N_NODES * HC;
        rgat_init<<<dim3((unsigned)((n + 255) / 256)), dim3(256), 0, stream>>>(
            out, bias, amax, denom);
    }
    // 1) T = x @ W   (WMMA f32 16x16x4, dual-tile dual-accumulator)
    rgat_gemm<<<dim3(N_NODES / 16, 4), dim3(32), 0, stream>>>(x, W, T);
    // 2) edge logits + segment max
    rgat_attn<<<dim3(N_EDGES / 8), dim3(256), 0, stream>>>(ei, et, T, q, kw,
                                                           alpha, amax);
    // 3) exp + segment sum
    rgat_softmax<<<dim3((N_EDGES * HEADS) / 256), dim3(256), 0, stream>>>(
        ei, amax, alpha, denom);
    // 4) weighted scatter-add aggregate
    rgat_agg<<<dim3(N_EDGES / 8), dim3(256), 0, stream>>>(ei, et, T, alpha,
                                                          denom, out);
}